// InterClassLoss_43258910605422
// MI455X (gfx1250) — compile-verified
//
#include <hip/hip_runtime.h>
#include <hip/hip_bf16.h>

typedef __attribute__((ext_vector_type(2))) float v2f;
typedef __attribute__((ext_vector_type(8))) float v8f;

#define C_ROWS 4096
#define DIM    1024
#define BM     64
#define BN     64
#define KC     32
#define NCHUNK (DIM / KC)
#define LDST   36          // KC + 4 pad: 16B-aligned float4 stores, conflict-free frag reads
#define MARGIN 0.0001f

// ---------------------------------------------------------------------------
// CDNA5 async global->LDS copy path (guarded so compile can't regress).
// ---------------------------------------------------------------------------
#if defined(__has_builtin)
#  if __has_builtin(__builtin_amdgcn_global_load_async_to_lds_b128)
#    define HAVE_ASYNC_LDS 1
#  endif
#endif
#ifndef HAVE_ASYNC_LDS
#  define HAVE_ASYNC_LDS 0
#endif

#if HAVE_ASYNC_LDS
typedef int v4i __attribute__((vector_size(4 * sizeof(int))));
typedef __attribute__((address_space(1))) v4i as1_v4i;   // HIP "__device__" (global)
typedef __attribute__((address_space(3))) v4i as3_v4i;   // HIP "__shared__" (LDS)

__device__ __forceinline__ void async_copy16(const float* g, float* l) {
    // emits: global_load_async_to_lds_b128 (tracked by ASYNCcnt)
    __builtin_amdgcn_global_load_async_to_lds_b128((as1_v4i*)g, (as3_v4i*)l, 0, 0);
}

template <int N>
__device__ __forceinline__ void wait_asynccnt() {
#  if __has_builtin(__builtin_amdgcn_s_wait_asynccnt)
    __builtin_amdgcn_s_wait_asynccnt(N);
#  else
    asm volatile("s_wait_asynccnt %0" :: "n"(N) : "memory");
#  endif
}
#endif

// Stage one K-chunk (A: rows i0.., B: rows j0..) into an LDS buffer.
// 128 threads x 4 passes x 16B per matrix = 64 rows x 32 floats each.
__device__ __forceinline__ void stage_chunk(float* __restrict__ As, float* __restrict__ Bs,
                                            const float* __restrict__ centers,
                                            int i0, int j0, int k0, int rowp, int c4) {
    #pragma unroll
    for (int p = 0; p < 4; ++p) {
        const int r = p * 16 + rowp;
        const float* ga = centers + (size_t)(i0 + r) * DIM + k0 + c4;
        const float* gb = centers + (size_t)(j0 + r) * DIM + k0 + c4;
        float* la = &As[r * LDST + c4];
        float* lb = &Bs[r * LDST + c4];
#if HAVE_ASYNC_LDS
        async_copy16(ga, la);
        async_copy16(gb, lb);
#else
        *(float4*)la = *(const float4*)ga;
        *(float4*)lb = *(const float4*)gb;
#endif
    }
}

// ---------------------------------------------------------------------------
// Kernel 1: row squared-norms -> d_ws ; also zero the scalar output.
// ---------------------------------------------------------------------------
__global__ __launch_bounds__(128) void row_norms_kernel(const float* __restrict__ centers,
                                                        float* __restrict__ sq,
                                                        float* __restrict__ out) {
    if (blockIdx.x == 0 && threadIdx.x == 0) out[0] = 0.0f;
    __shared__ float red[128];
    const int row = blockIdx.x;
    const float* p = centers + (size_t)row * DIM;
    float s = 0.f;
    for (int k = threadIdx.x; k < DIM; k += 128) {
        float v = p[k];
        s += v * v;
    }
    red[threadIdx.x] = s;
    __syncthreads();
    for (int off = 64; off > 0; off >>= 1) {
        if (threadIdx.x < off) red[threadIdx.x] += red[threadIdx.x + off];
        __syncthreads();
    }
    if (threadIdx.x == 0) sq[row] = red[0];
}

// ---------------------------------------------------------------------------
// Kernel 2: fused Gram-tile WMMA GEMM + hinge + triangular mask + reduction.
// 128 threads = 4 wave32s. Block tile 64x64, wave w owns rows [16w,16w+16).
// Double-buffered LDS, async DMA staging.
// ---------------------------------------------------------------------------
__global__ __launch_bounds__(128) void hinge_gemm_kernel(const float* __restrict__ centers,
                                                         const float* __restrict__ sq,
                                                         float* __restrict__ out) {
    const int bj = blockIdx.x;   // j (column) tile
    const int bi = blockIdx.y;   // i (row) tile
    if (bj < bi) return;         // strictly-lower tiles have no pairs with j > i

    __shared__ float As[2][BM * LDST];
    __shared__ float Bs[2][BN * LDST];
    __shared__ float red[128];

    const int t    = threadIdx.x;
    const int lane = t & 31;
    const int wave = t >> 5;
    const int i0   = bi * BM;
    const int j0   = bj * BN;

    v8f acc[4];
    #pragma unroll
    for (int n = 0; n < 4; ++n)
        #pragma unroll
        for (int r = 0; r < 8; ++r) acc[n][r] = 0.0f;

    // global->LDS staging assignment: 16 rows x (8 float4) per pass, 4 passes
    const int rowp = t >> 3;          // 0..15
    const int c4   = (t & 7) * 4;     // 0,4,...,28

    // WMMA fp32 fragment addressing (ISA 7.12.2, 32-bit A 16x4 layout):
    // lanes 0-15 hold K={kk,kk+1}, lanes 16-31 hold K={kk+2,kk+3}, M/N = lane%16
    const int lm = lane & 15;
    const int lk = (lane >> 4) * 2;   // 0 or 2

    // ---- software pipeline: prefetch chunk 0, then issue c+1 while computing c
    stage_chunk(As[0], Bs[0], centers, i0, j0, 0, rowp, c4);

    for (int c = 0; c < NCHUNK; ++c) {
        const int cur = c & 1;
        if (c + 1 < NCHUNK)
            stage_chunk(As[cur ^ 1], Bs[cur ^ 1], centers, i0, j0, (c + 1) * KC, rowp, c4);
#if HAVE_ASYNC_LDS
        // chunk c's 8 per-thread DMAs done; chunk c+1's 8 may remain in flight
        if (c + 1 < NCHUNK) wait_asynccnt<8>(); else wait_asynccnt<0>();
#endif
        __syncthreads();

        const float* Ab = As[cur];
        const float* Bb = Bs[cur];
        #pragma unroll
        for (int kk = 0; kk < KC; kk += 4) {
            const v2f a = *(const v2f*)(&Ab[(wave * 16 + lm) * LDST + kk + lk]);
            #pragma unroll
            for (int n = 0; n < 4; ++n) {
                const v2f b = *(const v2f*)(&Bb[(n * 16 + lm) * LDST + kk + lk]);
                acc[n] = __builtin_amdgcn_wmma_f32_16x16x4_f32(
                    /*neg_a=*/false, a, /*neg_b=*/false, b,
                    /*c_mod=*/(short)0, acc[n],
                    /*reuse_a=*/false, /*reuse_b=*/false);
            }
        }
        __syncthreads();   // all waves done reading buf `cur` before it is refilled
    }

    // Epilogue: d2 = |ci|^2 + |cj|^2 - 2*gram, hinge on strict upper triangle.
    // D layout: element r of acc -> M = r + 8*(lane/16), N = lane%16.
    const int mhalf = (lane >> 4) * 8;
    float sqi[8];
    #pragma unroll
    for (int r = 0; r < 8; ++r) sqi[r] = sq[i0 + wave * 16 + r + mhalf];

    float s = 0.f;
    #pragma unroll
    for (int n = 0; n < 4; ++n) {
        const int j     = j0 + n * 16 + lm;
        const float sqj = sq[j];
        #pragma unroll
        for (int r = 0; r < 8; ++r) {
            const int i = i0 + wave * 16 + r + mhalf;
            if (j > i) {
                float d2 = fmaxf(sqi[r] + sqj - 2.0f * acc[n][r], 0.0f);
                s += fmaxf(0.0f, MARGIN - sqrtf(d2));
            }
        }
    }

    red[t] = s;
    __syncthreads();
    for (int off = 64; off > 0; off >>= 1) {
        if (t < off) red[t] += red[t + off];
        __syncthreads();
    }
    if (t == 0) atomicAdd(out, red[0]);
}

// ---------------------------------------------------------------------------
extern "C" void kernel_launch(void* const* d_in, const int* in_sizes, int n_in,
                              void* d_out, int out_size, void* d_ws, size_t ws_size,
                              hipStream_t stream) {
    const float* centers = (const float*)d_in[0];
    float* out = (float*)d_out;
    float* sq  = (float*)d_ws;   // 4096 floats = 16 KB scratch

    row_norms_kernel<<<C_ROWS, 128, 0, stream>>>(centers, sq, out);

    dim3 grid(C_ROWS / BN, C_ROWS / BM);   // (bj, bi); bj < bi blocks exit immediately
    hinge_gemm_kernel<<<grid, 128, 0, stream>>>(centers, sq, out);
}